// CBAE_EndToEnd_70600672411689
// MI455X (gfx1250) — compile-verified
//
#include <hip/hip_runtime.h>

// CDNA5 / gfx1250 soft-rasterizer forward pass.
// s[edge,pixel] is affine in (gy,gx) -> computed with V_WMMA_F32_16X16X4_F32.
// Coverage = rcp( prod_k (1 + exp(-s_k)) ), composited back-to-front.

typedef __attribute__((ext_vector_type(2))) float v2f;
typedef __attribute__((ext_vector_type(8))) float v8f;

#define HH 128
#define WW 128
#define NN 128
#define KK 12
#define STATE (NN * KK * 2 + NN)   // 3200 floats per frame
#define INV_SOFT 100.0f            // 1 / fallback_softness

__global__ __launch_bounds__(256) void raster_wmma_kernel(
    const float* __restrict__ traj,    // [T, 3200]
    const float* __restrict__ colors,  // [1, 128, 3]
    const float* __restrict__ alpha,   // [128]
    const float* __restrict__ zbuf,    // [128]
    const int*   __restrict__ csg,     // [128] (bool -> int)
    float* __restrict__ out)           // [1, T, 128, 128, 3]
{
    // A-operand staging: sAB[prim][lane] = {vgpr0, vgpr1} for WMMA A (16x4 f32):
    //   lanes 0..15  -> edge=lane,  {K0=coefY, K1=coefX}
    //   lanes 16..31 -> edge=lane-16,{K2=coefC, K3=0}
    __shared__ float2 sAB[NN * 32];
    __shared__ float4 sC4[NN];   // {r*(1-csg), g*(1-csg), b*(1-csg), alpha*sigmoid(alive)}
    __shared__ float  sZ[NN];

    const int t   = blockIdx.y;
    const int row = blockIdx.x;
    const int tid = threadIdx.x;
    const float* st = traj + (size_t)t * STATE;

    if (tid < NN) sZ[tid] = zbuf[tid];
    __syncthreads();

    if (tid < NN) {
        const int n = tid;
        // stable ascending rank == position in jnp.argsort(z)
        const float zn = sZ[n];
        int rank = 0;
        for (int j = 0; j < NN; ++j) {
            float zj = sZ[j];
            rank += (zj < zn) || ((zj == zn) && (j < n)) ? 1 : 0;
        }
        // per-primitive alpha weight and CSG factor
        const float alive = st[NN * KK * 2 + n];
        const float aw = alpha[n] * __builtin_amdgcn_rcpf(1.0f + __expf(-alive));
        const float keep = csg[n] ? 0.0f : 1.0f;   // (1 - sub)
        sC4[rank] = make_float4(colors[n * 3 + 0] * keep,
                                colors[n * 3 + 1] * keep,
                                colors[n * 3 + 2] * keep, aw);
        // edge coefficients, orientation folded in with 1/SOFT scale
        const float* P = st + n * (KK * 2);
        float area2 = 0.0f;
        for (int k = 0; k < KK; ++k) {
            int k1 = (k == KK - 1) ? 0 : k + 1;
            area2 += P[2 * k] * P[2 * k1 + 1] - P[2 * k1] * P[2 * k + 1];
        }
        const float orient = (area2 > 0.0f) ? 1.0f : ((area2 < 0.0f) ? -1.0f : 0.0f);
        const float sc = orient * INV_SOFT;
        for (int k = 0; k < KK; ++k) {
            int k1 = (k == KK - 1) ? 0 : k + 1;
            float x0 = P[2 * k], y0 = P[2 * k + 1];
            float ex = P[2 * k1] - x0;
            float ey = P[2 * k1 + 1] - y0;
            // s = (sc*ex)*gy + (-sc*ey)*gx + sc*(ey*x0 - ex*y0)
            sAB[rank * 32 + k]      = make_float2(sc * ex, -sc * ey);
            sAB[rank * 32 + 16 + k] = make_float2(sc * (ey * x0 - ex * y0), 0.0f);
        }
        for (int k = KK; k < 16; ++k) {   // pad edges: s = 3e4 -> factor (1+e^-s) == 1
            sAB[rank * 32 + k]      = make_float2(0.0f, 0.0f);
            sAB[rank * 32 + 16 + k] = make_float2(3.0e4f, 0.0f);
        }
    }
    __syncthreads();

    // ---- main rasterization: one wave = 16 pixels of this row ----
    const int lane = tid & 31;
    const int col  = lane & 15;
    const int wave = tid >> 5;
    const int x    = wave * 16 + col;
    const float gx = ((float)x   + 0.5f) * (1.0f / WW);
    const float gy = ((float)row + 0.5f) * (1.0f / HH);

    // B operand (4x16 f32): rows K0=gy, K1=gx, K2=1, K3=0; column = pixel
    v2f bm;
    bm.x = (lane < 16) ? gy : 1.0f;   // vgpr0: K0 (lanes<16) / K2 (lanes>=16)
    bm.y = (lane < 16) ? gx : 0.0f;   // vgpr1: K1 (lanes<16) / K3 (lanes>=16)
    const v8f czero = {};

    float Tacc = 1.0f;
    float accR = 0.0f, accG = 0.0f, accB = 0.0f;

    // back-to-front: iterate sorted primitives from far end, keep exclusive transmittance
    for (int i = NN - 1; i >= 0; --i) {
        float2 ab = sAB[i * 32 + lane];           // ds_load_b64, conflict-free
        v2f am; am.x = ab.x; am.y = ab.y;
        v8f s = __builtin_amdgcn_wmma_f32_16x16x4_f32(
            false, am, false, bm, (short)0, czero, false, false);
        // denominator product over this lane's 8 edges (factors are all >= 1)
        float prod =  (1.0f + __expf(-s[0]));
        prod      *=  (1.0f + __expf(-s[1]));
        prod      *=  (1.0f + __expf(-s[2]));
        prod      *=  (1.0f + __expf(-s[3]));
        prod      *=  (1.0f + __expf(-s[4]));
        prod      *=  (1.0f + __expf(-s[5]));
        prod      *=  (1.0f + __expf(-s[6]));
        prod      *=  (1.0f + __expf(-s[7]));
        float other = __shfl_xor(prod, 16, 32);   // merge edges 0-7 with 8-15
        float cov = __builtin_amdgcn_rcpf(prod * other);   // in [0,1]
        float4 cc = sC4[i];
        float a   = cov * cc.w;
        float fac = a * Tacc;
        accR += fac * cc.x;
        accG += fac * cc.y;
        accB += fac * cc.z;
        Tacc *= (1.0f - a);
    }

    if (lane < 16) {
        size_t o = (((size_t)t * HH + row) * WW + x) * 3;
        out[o + 0] = accR;
        out[o + 1] = accG;
        out[o + 2] = accB;
    }
}

extern "C" void kernel_launch(void* const* d_in, const int* in_sizes, int n_in,
                              void* d_out, int out_size, void* d_ws, size_t ws_size,
                              hipStream_t stream) {
    const float* traj   = (const float*)d_in[0];
    const float* colors = (const float*)d_in[1];
    const float* alpha  = (const float*)d_in[2];
    const float* zbuf   = (const float*)d_in[3];
    const int*   csg    = (const int*)d_in[4];
    float* out = (float*)d_out;

    const int T = in_sizes[0] / STATE;   // 192
    dim3 grid(HH, T);                    // one block = one image row of one frame
    raster_wmma_kernel<<<grid, 256, 0, stream>>>(traj, colors, alpha, zbuf, csg, out);
}